// Attention_19971597927194
// MI455X (gfx1250) — compile-verified
//
#include <hip/hip_runtime.h>
#include <math.h>

typedef __attribute__((ext_vector_type(2))) float v2f;
typedef __attribute__((ext_vector_type(8))) float v8f;

#define HD 128
#define ED 32

// ---------------------------------------------------------------------------
// Phase 1: per-edge scores via V_WMMA_F32_16X16X4_F32 diagonal trick.
// One wave32 handles 16 edges. A = q rows of the 16 edges (16xK), B = k rows
// as columns (Kx16). D = A*B^? -> we only consume the diagonal D[j][j], which
// is exactly dot(q[i0[e_j]], k[i1[e_j]]) in full f32 precision.
//
// Documented layouts (CDNA5 ISA):
//   A 16x4 f32 : lanes 0-15 -> M=0..15, lanes 16-31 -> M=0..15 again;
//                VGPR0 holds K={0 | 2}, VGPR1 holds K={1 | 3} (low|high half)
//   B 4x16 f32 : lanes carry N, same (VGPR, half)->K striping
//   D 16x16 f32: VGPR r: lanes 0-15 -> (M=r, N=lane); lanes 16-31 -> (M=r+8)
// => edge j lives in lanes {j, j+16}; diag j<8 at lane j / VGPR j,
//    diag j>=8 at lane j+16 / VGPR j-8.  Owning lanes: l<8 or l>=24.
// ---------------------------------------------------------------------------
__global__ void edge_scores_kernel(
    const float* __restrict__ q, const float* __restrict__ k,
    const float* __restrict__ eigs, const float* __restrict__ lambda0,
    const float* __restrict__ path_emb,
    const int* __restrict__ i0, const int* __restrict__ i1,
    const int* __restrict__ path_type,
    float* __restrict__ w0, float* __restrict__ w1,
    float* __restrict__ sum0, float* __restrict__ sum1,
    int E)
{
    const int lane = (int)(threadIdx.x & 31u);
    const int wave = (int)((blockIdx.x * blockDim.x + threadIdx.x) >> 5);
    const long base = (long)wave * 16;
    if (base >= E) return;                       // wave-uniform exit

    const int  j    = lane & 15;                 // edge slot within group
    long e = base + j;
    const bool live = e < (long)E;
    if (!live) e = (long)E - 1;                  // clamp loads for ragged tail

    const int r0 = i0[e];
    const int r1 = i1[e];
    const int half2 = (lane >> 4) * 2;           // K sub-offset: 0 or 2

    const float* qr = q + (long)r0 * HD;
    const float* kr = k + (long)r1 * HD;

    v8f cx = {0.f,0.f,0.f,0.f,0.f,0.f,0.f,0.f};
#pragma unroll
    for (int s = 0; s < HD / 4; ++s) {
        const int off = 4 * s + half2;
        v2f a = *reinterpret_cast<const v2f*>(qr + off);   // 8B aligned
        v2f b = *reinterpret_cast<const v2f*>(kr + off);
        cx = __builtin_amdgcn_wmma_f32_16x16x4_f32(
                 false, a, false, b, (short)0, cx, false, false);
    }

    const float* er0 = eigs + (long)r0 * ED;
    const float* er1 = eigs + (long)r1 * ED;
    v8f cy = {0.f,0.f,0.f,0.f,0.f,0.f,0.f,0.f};
#pragma unroll
    for (int s = 0; s < ED / 4; ++s) {
        const int off = 4 * s + half2;
        v2f a = *reinterpret_cast<const v2f*>(er0 + off);
        v2f b = *reinterpret_cast<const v2f*>(er1 + off);
        cy = __builtin_amdgcn_wmma_f32_16x16x4_f32(
                 false, a, false, b, (short)0, cy, false, false);
    }

    // Harvest diagonal: owning lanes are l<8 (elem l) and l>=24 (elem l-24).
    const bool own = (lane < 8) || (lane >= 24);
    const int  d   = (lane < 8) ? lane : (lane - 24);
    float xv = cx[0];
    float yv = cy[0];
#pragma unroll
    for (int t = 1; t < 8; ++t) {
        xv = (d == t) ? cx[t] : xv;
        yv = (d == t) ? cy[t] : yv;
    }

    if (own && live) {
        const float lam = expf(lambda0[0]);
        const float x   = xv * 0.08838834764831845f;       // 1/sqrt(128)
        // jnp.clip(exp(v), -5, 5) == min(exp(v), 5) since exp >= 0
        const float sc0 = fminf(expf(x + lam * yv), 5.0f);
        w0[e] = sc0;
        atomicAdd(&sum0[r0], sc0);
        const int pt = path_type[e];
        const float sc1 = fminf(expf(path_emb[pt]), 5.0f);
        w1[e] = sc1;
        atomicAdd(&sum1[r0], sc1);
    }
}

// ---------------------------------------------------------------------------
// Phase 2: normalize both channels, average, and scatter s * v[i1] into
// out[i0]. One wave per edge; 32 lanes x float4 = 128 floats.
// ---------------------------------------------------------------------------
__global__ void scatter_kernel(
    const float* __restrict__ v,
    const int* __restrict__ i0, const int* __restrict__ i1,
    const float* __restrict__ w0, const float* __restrict__ w1,
    const float* __restrict__ sum0, const float* __restrict__ sum1,
    float* __restrict__ out, int E)
{
    const int  lane = (int)(threadIdx.x & 31u);
    const long wid  = ((long)blockIdx.x * blockDim.x + threadIdx.x) >> 5;
    if (wid >= (long)E) return;                  // wave-uniform exit

    const int r0 = i0[wid];
    const int r1 = i1[wid];
    float s0 = sum0[r0]; s0 = (s0 == 0.0f) ? 1.0f : s0;
    float s1 = sum1[r0]; s1 = (s1 == 0.0f) ? 1.0f : s1;
    const float s = 0.5f * (w0[wid] / s0 + w1[wid] / s1);

    const float4 val =
        *reinterpret_cast<const float4*>(v + (long)r1 * HD + lane * 4);
    float* orow = out + (long)r0 * HD + lane * 4;
    atomicAdd(orow + 0, s * val.x);
    atomicAdd(orow + 1, s * val.y);
    atomicAdd(orow + 2, s * val.z);
    atomicAdd(orow + 3, s * val.w);
}

// ---------------------------------------------------------------------------
// inputs (setup_inputs order): q, k, v, eigs, lambda0, path_emb, indices,
// path_type.  indices is (2,E) flat: i0 = idx[0:E), i1 = idx[E:2E).
// workspace: sum0[N] | sum1[N] | w0[E] | w1[E]  (floats)
// ---------------------------------------------------------------------------
extern "C" void kernel_launch(void* const* d_in, const int* in_sizes, int n_in,
                              void* d_out, int out_size, void* d_ws, size_t ws_size,
                              hipStream_t stream) {
    const float* q        = (const float*)d_in[0];
    const float* k        = (const float*)d_in[1];
    const float* v        = (const float*)d_in[2];
    const float* eigs     = (const float*)d_in[3];
    const float* lambda0  = (const float*)d_in[4];
    const float* path_emb = (const float*)d_in[5];
    const int*   idx      = (const int*)d_in[6];
    const int*   ptype    = (const int*)d_in[7];

    const int N = in_sizes[0] / HD;
    const int E = in_sizes[7];
    const int* i0 = idx;
    const int* i1 = idx + E;

    float* ws   = (float*)d_ws;
    float* sum0 = ws;
    float* sum1 = ws + N;
    float* w0   = ws + 2L * N;
    float* w1   = w0 + E;
    float* out  = (float*)d_out;

    // zero the segment sums and the output accumulator (capture-safe memsets)
    hipMemsetAsync(sum0, 0, (size_t)(2L * N) * sizeof(float), stream);
    hipMemsetAsync(out,  0, (size_t)out_size * sizeof(float), stream);

    // phase 1: 1 wave per 16 edges, 4 waves per 128-thread block
    {
        const long waves  = ((long)E + 15) / 16;
        const long blocks = (waves + 3) / 4;
        edge_scores_kernel<<<(unsigned)blocks, 128, 0, stream>>>(
            q, k, eigs, lambda0, path_emb, i0, i1, ptype,
            w0, w1, sum0, sum1, E);
    }

    // phase 2: 1 wave per edge, 8 waves per 256-thread block
    {
        const long blocks = ((long)E + 7) / 8;
        scatter_kernel<<<(unsigned)blocks, 256, 0, stream>>>(
            v, i0, i1, w0, w1, sum0, sum1, out, E);
    }
}